// PchipF0UpsamplerTorch_15676630630852
// MI455X (gfx1250) — compile-verified
//
#include <hip/hip_runtime.h>

typedef __attribute__((ext_vector_type(2))) float v2f;
typedef __attribute__((ext_vector_type(8))) float v8f;

#define NB 32
#define T 2048
#define UPSCALE 480
#define T_OUT (T * UPSCALE) /* 983040 */

__device__ __forceinline__ float signf(float v) {
    return (v > 0.f) ? 1.f : ((v < 0.f) ? -1.f : 0.f);
}

__device__ __forceinline__ float pchip_edge(float dA, float dB) {
    float e = (3.f * dA - dB) * 0.5f;
    if (signf(e) != signf(dA)) e = 0.f;
    if ((signf(dA) != signf(dB)) && (fabsf(e) > 3.f * fabsf(dA))) e = 3.f * dA;
    return e;
}

// Kernel 1: Fritsch-Carlson monotone slopes, one thread per (b,t). 256 KB out.
__global__ void pchip_slopes_kernel(const float* __restrict__ x,
                                    float* __restrict__ d) {
    int gid = blockIdx.x * blockDim.x + threadIdx.x;
    if (gid >= NB * T) return;
    int b = gid / T, t = gid - b * T;
    const float* y = x + b * T;
    float dv;
    if (t == 0) {
        dv = pchip_edge(y[1] - y[0], y[2] - y[1]);
    } else if (t == T - 1) {
        dv = pchip_edge(y[T - 1] - y[T - 2], y[T - 2] - y[T - 3]);
    } else {
        float d0 = y[t] - y[t - 1];
        float d1 = y[t + 1] - y[t];
        float prod = d0 * d1;
        dv = (prod > 0.f) ? (2.f * prod / (d0 + d1)) : 0.f;
    }
    d[gid] = dv;
}

// Kernel 2: Hermite evaluation as exact K=4 WMMA contraction.
// Tile: M=16 batches x N=16 consecutive t_out. Two v_wmma_f32_16x16x4_f32 per
// tile (basis rows masked per knot interval i0 / i0+1), accumulated in C.
__global__ void pchip_upsample_wmma_kernel(const float* __restrict__ x,
                                           const float* __restrict__ dp,
                                           float* __restrict__ out) {
    const double step = (double)(T - 1) / (double)(T_OUT - 1);
    const int lane = threadIdx.x & 31;
    const int l16 = lane & 15;
    const bool hi = lane >= 16;

    int wave = (blockIdx.x * blockDim.x + threadIdx.x) >> 5;
    int nwaves = (gridDim.x * blockDim.x) >> 5;
    const int ntilesN = T_OUT / 16;     // 61440 time tiles
    const int ntiles = ntilesN * 2;     // x2 batch tiles (32 batches)

    for (int tile = wave; tile < ntiles; tile += nwaves) {
        int bt = tile & 1;      // batch half
        int nt = tile >> 1;     // time tile
        int n0 = nt << 4;
        int bb = bt << 4;

        // Knot intervals covered by this tile: {i0, i1}, i1 clamped.
        int i0 = (int)((double)n0 * step);
        if (i0 > T - 2) i0 = T - 2;
        int i1 = (i0 < T - 2) ? (i0 + 1) : (T - 2);

        // ---- A matrix (16x4 f32): row m = batch bb+l16.
        // lanes 0-15 : v0=K0=y[i],   v1=K1=d[i]
        // lanes 16-31: v0=K2=y[i+1], v1=K3=d[i+1]   (i0+1 <= 2047: in range)
        const float* yb = x + (bb + l16) * T;
        const float* db = dp + (bb + l16) * T;
        int ia1 = hi ? (i0 + 1) : i0;
        int ia2 = hi ? (i1 + 1) : i1;
        v2f a1, a2;
        a1.x = yb[ia1];  a1.y = db[ia1];
        a2.x = yb[ia2];  a2.y = db[ia2];

        // ---- B matrix (4x16 f32): col n = n0+l16 = t_out; Hermite basis.
        int n = n0 + l16;
        double t = (double)n * step;
        int idxn = (int)t;
        if (idxn > T - 2) idxn = T - 2;
        float s = (float)(t - (double)idxn);
        float oms = 1.f - s;
        float s2 = s * s;
        float h00 = (1.f + 2.f * s) * oms * oms;
        float h10 = s * oms * oms;
        float h01 = s2 * (3.f - 2.f * s);
        float h11 = s2 * (s - 1.f);
        // lanes 0-15: v0=K0->h00, v1=K1->h10 ; lanes 16-31: v0=K2->h01, v1=K3->h11
        float bk0 = hi ? h01 : h00;
        float bk1 = hi ? h11 : h10;
        bool in0 = (idxn == i0);        // rows split exactly between i0 / i1
        v2f b1, b2;
        b1.x = in0 ? bk0 : 0.f;  b1.y = in0 ? bk1 : 0.f;
        b2.x = in0 ? 0.f : bk0;  b2.y = in0 ? 0.f : bk1;

        v8f c = {};
        c = __builtin_amdgcn_wmma_f32_16x16x4_f32(false, a1, false, b1,
                                                  (short)0, c, false, false);
        c = __builtin_amdgcn_wmma_f32_16x16x4_f32(false, a2, false, b2,
                                                  (short)0, c, false, false);

        // ---- D store: VGPR v -> batch bb+v (lanes 0-15) / bb+v+8 (lanes 16-31),
        // N = l16 -> column n0+l16. Each store = two contiguous 64B runs.
        int outb = bb + (hi ? 8 : 0);
        size_t col = (size_t)n0 + (size_t)l16;
        float* orow = out + (size_t)outb * (size_t)T_OUT + col;
#pragma unroll
        for (int v = 0; v < 8; ++v) {
            orow[(size_t)v * (size_t)T_OUT] = c[v];
        }
    }
}

extern "C" void kernel_launch(void* const* d_in, const int* in_sizes, int n_in,
                              void* d_out, int out_size, void* d_ws, size_t ws_size,
                              hipStream_t stream) {
    (void)in_sizes; (void)n_in; (void)out_size; (void)ws_size;
    const float* x = (const float*)d_in[0];
    float* slopes = (float*)d_ws;            // 32*2048*4 = 256 KB scratch
    float* out = (float*)d_out;

    // Kernel 1: slopes (65536 threads)
    pchip_slopes_kernel<<<(NB * T + 255) / 256, 256, 0, stream>>>(x, slopes);

    // Kernel 2: one 16x16 tile per wave32; 61440*2 tiles -> 15360 blocks of 256.
    const int ntiles = (T_OUT / 16) * 2;
    const int waves_per_block = 256 / 32;
    int blocks = (ntiles + waves_per_block - 1) / waves_per_block;
    pchip_upsample_wmma_kernel<<<blocks, 256, 0, stream>>>(x, slopes, out);
}